// SecondOrderFeatureInteraction_18957985644948
// MI455X (gfx1250) — compile-verified
//
#include <hip/hip_runtime.h>
#include <hip/hip_bf16.h>

typedef __attribute__((ext_vector_type(2))) float v2f;
typedef __attribute__((ext_vector_type(8))) float v8f;

#define FEAS 27
#define DIM  128
#define NPAIR 351   // 27*26/2
#define WAVES_PER_BLOCK 8

__global__ __launch_bounds__(256) void soi_wmma_f32_kernel(
    const float* __restrict__ X,   // [B, 27, 128]
    float* __restrict__ out,       // [B, 351]
    int B)
{
    const int lane = threadIdx.x & 31;
    const int wave = threadIdx.x >> 5;
    const int b = blockIdx.x * WAVES_PER_BLOCK + wave;
    if (b >= B) return;   // wave-uniform

    const int half = lane >> 4;   // 0: K pair {0,1}; 1: K pair {2,3} within each K=4 step
    const int r    = lane & 15;   // row (A-operand) == col (B-operand) index

    // Rows 27..31 of the padded 32-row block don't exist in memory. Instead of
    // zero-filling (which costs per-iteration EXEC masking), clamp those lanes
    // to an in-bounds row of the SAME batch. The resulting garbage only lands in
    // Gram elements with tile-row M>=11 or tile-col N>=11 of T10/T11, all of
    // which are discarded by the store conditions below. No OOB, no divergence.
    const int row1 = (16 + r) < FEAS ? (16 + r) : r;

    const float* base = X + (size_t)b * FEAS * DIM;
    const float* p0 = base + r * DIM + half * 2;      // rows 0..15
    const float* p1 = base + row1 * DIM + half * 2;   // rows 16..26 (clamped)

    v8f c00 = {}; // G[0:16, 0:16]
    v8f c10 = {}; // G[16:32, 0:16]
    v8f c11 = {}; // G[16:32,16:32]

#pragma unroll
    for (int kk = 0; kk < DIM / 4; ++kk) {
        // A-operand layout == B-operand layout for a Gram product:
        // lane holds X[row = lane%16][k = 4*kk + 2*half + {0,1}]
        v2f a0 = *(const v2f*)(p0 + kk * 4);
        v2f a1 = *(const v2f*)(p1 + kk * 4);

        // D = A(16x4) * B(4x16) + C ; fp32 exact MAC path
        c00 = __builtin_amdgcn_wmma_f32_16x16x4_f32(false, a0, false, a0,
                                                    (short)0, c00, false, false);
        c10 = __builtin_amdgcn_wmma_f32_16x16x4_f32(false, a1, false, a0,
                                                    (short)0, c10, false, false);
        c11 = __builtin_amdgcn_wmma_f32_16x16x4_f32(false, a1, false, a1,
                                                    (short)0, c11, false, false);
    }

    // C/D layout: VGPR v, lanes 0-15 -> M=v, lanes 16-31 -> M=v+8; N = lane%16.
    // Output pair (row R, col C), R > C, index = R*(R-1)/2 + C.
    float* ob = out + (size_t)b * NPAIR;
#pragma unroll
    for (int v = 0; v < 8; ++v) {
        const int M = v + half * 8;  // tile-local row
        const int N = r;             // tile-local col

        // tile00: R = M, C = N, need M > N
        if (M > N)
            ob[(M * (M - 1)) / 2 + N] = c00[v];

        // tile10: R = 16+M, C = N; need R <= 26 (always R > C here)
        if (M <= 10)
            ob[((16 + M) * (15 + M)) / 2 + N] = c10[v];

        // tile11: R = 16+M, C = 16+N; need R <= 26 and N < M
        if (M <= 10 && N < M)
            ob[((16 + M) * (15 + M)) / 2 + 16 + N] = c11[v];
    }
}

extern "C" void kernel_launch(void* const* d_in, const int* in_sizes, int n_in,
                              void* d_out, int out_size, void* d_ws, size_t ws_size,
                              hipStream_t stream) {
    const float* X = (const float*)d_in[0];
    float* out = (float*)d_out;
    const int B = in_sizes[0] / (FEAS * DIM);   // 16384

    const int blocks = (B + WAVES_PER_BLOCK - 1) / WAVES_PER_BLOCK;
    soi_wmma_f32_kernel<<<blocks, 256, 0, stream>>>(X, out, B);
}